// CMA_87625922773344
// MI455X (gfx1250) — compile-verified
//
#include <hip/hip_runtime.h>
#include <stdint.h>

typedef float v2f __attribute__((ext_vector_type(2)));
typedef float v8f __attribute__((ext_vector_type(8)));

#define D_DIM   2048
#define CBINS   4096      // padded class bins (C=4000), 256 threads * 16
#define ROW_F   2056      // 2048 + 8 pad floats: rows 2 apart shift 16 banks
#define CH_ROWS 8         // rows staged per chunk (two K=4 WMMA steps)
#define EPSN    1e-12f
#define MOM     0.9f

// ---------------- CDNA5 async global->LDS staging ----------------
__device__ __forceinline__ void async_ld_b128(const float* g, const float* l) {
  uint32_t lds = (uint32_t)(uintptr_t)l;          // low 32 bits = LDS offset
  uint64_t ga  = (uint64_t)(uintptr_t)g;
  asm volatile("global_load_async_to_lds_b128 %0, %1, off"
               :: "v"(lds), "v"(ga) : "memory");
}

// ---------------- utility kernels ----------------
__global__ void k_zero_i32(int* p, int n) {
  int i = blockIdx.x * blockDim.x + threadIdx.x;
  if (i < n) p[i] = 0;
}

// one wave per feature row: row L2 norm + class histogram
__global__ __launch_bounds__(256) void k_rownorm(
    const float* __restrict__ f0, const float* __restrict__ f1,
    const int* __restrict__ i0, const int* __restrict__ i1,
    float* __restrict__ n0, float* __restrict__ n1,
    int* __restrict__ c0, int* __restrict__ c1, int N)
{
  const int mod = blockIdx.y;
  const float* f   = mod ? f1 : f0;
  const int*  ids  = mod ? i1 : i0;
  float*      invn = mod ? n1 : n0;
  int*        cnts = mod ? c1 : c0;
  const int wv = threadIdx.x >> 5, lane = threadIdx.x & 31;
  const int row = blockIdx.x * 8 + wv;
  if (row >= N) return;
  const float4* rp = (const float4*)(f + (size_t)row * D_DIM);
  float ssq = 0.0f;
#pragma unroll
  for (int j = 0; j < 16; ++j) {          // 16 * 32 lanes * float4 = 2048
    float4 v = rp[j * 32 + lane];
    ssq += v.x * v.x + v.y * v.y + v.z * v.z + v.w * v.w;
  }
#pragma unroll
  for (int m = 16; m >= 1; m >>= 1) ssq += __shfl_xor(ssq, m, 32);
  if (lane == 0) {
    invn[row] = 1.0f / fmaxf(sqrtf(ssq), EPSN);
    atomicAdd(&cnts[ids[row]], 1);
  }
}

// single-block exclusive scan of the 4096-bin histogram -> offsets + cursors
__global__ __launch_bounds__(256) void k_scan(
    const int* __restrict__ c0, const int* __restrict__ c1,
    int* __restrict__ o0, int* __restrict__ o1,
    int* __restrict__ u0, int* __restrict__ u1)
{
  const int mod = blockIdx.y;
  const int* cnts = mod ? c1 : c0;
  int* offs = mod ? o1 : o0;
  int* cur  = mod ? u1 : u0;
  __shared__ int s[256];
  const int tid = threadIdx.x;
  int loc[16]; int sum = 0;
#pragma unroll
  for (int j = 0; j < 16; ++j) { loc[j] = cnts[tid * 16 + j]; sum += loc[j]; }
  s[tid] = sum; __syncthreads();
  for (int off = 1; off < 256; off <<= 1) {
    int v = (tid >= off) ? s[tid - off] : 0;
    __syncthreads();
    s[tid] += v;
    __syncthreads();
  }
  int run = s[tid] - sum;                 // exclusive base
#pragma unroll
  for (int j = 0; j < 16; ++j) {
    offs[tid * 16 + j] = run; cur[tid * 16 + j] = run; run += loc[j];
  }
}

// build permutation (rows grouped by class) and fused weight w = invnorm/count
__global__ __launch_bounds__(256) void k_scatter(
    const int* __restrict__ i0, const int* __restrict__ i1,
    const float* __restrict__ n0, const float* __restrict__ n1,
    const int* __restrict__ c0, const int* __restrict__ c1,
    int* __restrict__ u0, int* __restrict__ u1,
    int* __restrict__ p0, int* __restrict__ p1,
    float* __restrict__ w0, float* __restrict__ w1, int N)
{
  const int mod = blockIdx.y;
  const int*   ids  = mod ? i1 : i0;
  const float* invn = mod ? n1 : n0;
  const int*   cnts = mod ? c1 : c0;
  int*   cur  = mod ? u1 : u0;
  int*   perm = mod ? p1 : p0;
  float* wr   = mod ? w1 : w0;
  const int i = blockIdx.x * 256 + threadIdx.x;
  if (i < N) {
    const int c = ids[i];
    const int pos = atomicAdd(&cur[c], 1);
    perm[pos] = i;
    wr[i] = invn[i] / (float)cnts[c];
  }
}

// ---------------- main WMMA segment-mean + EMA kernel ----------------
// One workgroup (8 waves) per group of 16 classes. Wave w owns cols [256w,256w+256).
// acc[t] is a 16x16 f32 C/D tile (classes x cols) for col-tile t.
// Double-buffered async global->LDS staging: chunk c+1 DMA overlaps chunk c WMMAs.
__global__ __launch_bounds__(256) void k_segmean_wmma(
    const float* __restrict__ feat, const int* __restrict__ ids,
    const int* __restrict__ perm, const float* __restrict__ wrow,
    const int* __restrict__ offsets, const int* __restrict__ counts,
    const float* __restrict__ mem, float* __restrict__ outp)
{
  __shared__ float s_rows[2][CH_ROWS * ROW_F];   // ~132KB ping-pong row stage
  __shared__ float s_red[16];
  __shared__ float s_inv[16];
  __shared__ int   s_pres[16];

  const int g     = blockIdx.x;
  const int tid   = threadIdx.x;
  const int wv    = tid >> 5;
  const int lane  = tid & 31;
  const int lhalf = lane & 15;        // N-index / M-in-half
  const int hi    = lane >> 4;        // half select
  const int wcol  = wv * 256;

  const int rbeg = offsets[g * 16];
  const int rend = offsets[g * 16 + 16];   // offsets padded to 4096 -> always valid

  v8f acc[16];
#pragma unroll
  for (int t = 0; t < 16; ++t)
#pragma unroll
    for (int i = 0; i < 8; ++i) acc[t][i] = 0.0f;

  // per-chunk metadata (uniform addresses -> scalarizable loads, no LDS)
  int   rcur[CH_ROWS],  rnxt[CH_ROWS];
  int   lcur[CH_ROWS],  lnxt[CH_ROWS];
  float wcur[CH_ROWS],  wnxt[CH_ROWS];

#define LOAD_META(BASE, RA, LA, WA)                                  \
  {                                                                  \
    _Pragma("unroll")                                                \
    for (int j = 0; j < CH_ROWS; ++j) {                              \
      const int k = (BASE) + j;                                      \
      const bool valid = (k < rend);                                 \
      const int r = valid ? perm[k] : 0;                             \
      RA[j] = r;                                                     \
      LA[j] = valid ? (ids[r] - g * 16) : -1;                        \
      WA[j] = valid ? wrow[r] : 0.0f;                                \
    }                                                                \
  }

#define ISSUE(RA, BUF)                                               \
  {                                                                  \
    _Pragma("unroll")                                                \
    for (int it = 0; it < 16; ++it) {                                \
      const int idx = it * 256 + tid;  /* 0..4095 */                 \
      const int rr  = idx >> 9;        /* /512 f4-per-row */         \
      const int fc  = idx & 511;                                     \
      async_ld_b128(feat + (size_t)RA[rr] * D_DIM + fc * 4,          \
                    &s_rows[BUF][rr * ROW_F + fc * 4]);              \
    }                                                                \
  }

  if (rbeg < rend) {
    LOAD_META(rbeg, rcur, lcur, wcur);
    ISSUE(rcur, 0);
  }

  int cb = 0;
  for (int base = rbeg; base < rend; base += CH_ROWS) {
    const int  nb      = cb ^ 1;
    const bool hasnext = (base + CH_ROWS) < rend;      // uniform
    if (hasnext) LOAD_META(base + CH_ROWS, rnxt, lnxt, wnxt);
    __syncthreads();                 // all waves done computing chunk c-1 (buffer nb)
    if (hasnext) {
      ISSUE(rnxt, nb);               // overlap next chunk's DMA with this compute
      asm volatile("s_wait_asynccnt 16" ::: "memory");   // retire chunk c only
    } else {
      asm volatile("s_wait_asynccnt 0" ::: "memory");
    }
    __syncthreads();                 // chunk c rows visible to all waves

#pragma unroll
    for (int k4 = 0; k4 < 2; ++k4) {
      // A (16x4): lanes0-15 hold K=0,1 ; lanes16-31 hold K=2,3
      const int r0 = k4 * 4 + hi * 2, r1 = r0 + 1;
      v2f a;
      a.x = (lcur[r0] == lhalf) ? wcur[r0] : 0.0f;
      a.y = (lcur[r1] == lhalf) ? wcur[r1] : 0.0f;
      const float* rp0 = &s_rows[cb][r0 * ROW_F + wcol];
      const float* rp1 = &s_rows[cb][r1 * ROW_F + wcol];
#pragma unroll
      for (int t = 0; t < 16; ++t) {
        v2f b;                       // B (4x16) mirrored layout
        b.x = rp0[t * 16 + lhalf];
        b.y = rp1[t * 16 + lhalf];
        acc[t] = __builtin_amdgcn_wmma_f32_16x16x4_f32(
            false, a, false, b, (short)0, acc[t], false, false);
      }
    }

    if (hasnext) {
#pragma unroll
      for (int j = 0; j < CH_ROWS; ++j) {
        rcur[j] = rnxt[j]; lcur[j] = lnxt[j]; wcur[j] = wnxt[j];
      }
    }
    cb = nb;
  }
#undef LOAD_META
#undef ISSUE

  // ---- epilogue: acc[t] = per-class normalized mean tile ----
  __syncthreads();
  if (tid < 16) { s_red[tid] = 0.0f; s_pres[tid] = (counts[g * 16 + tid] > 0) ? 1 : 0; }
  __syncthreads();

  // ||mean||^2 per class: C-layout M = v + 8*hi, N = lhalf
  float p[8];
#pragma unroll
  for (int v = 0; v < 8; ++v) {
    float s = 0.0f;
#pragma unroll
    for (int t = 0; t < 16; ++t) { const float x = acc[t][v]; s += x * x; }
    p[v] = s;
  }
#pragma unroll
  for (int m = 1; m < 16; m <<= 1)
#pragma unroll
    for (int v = 0; v < 8; ++v) p[v] += __shfl_xor(p[v], m, 32);
  if (lhalf == 0)
#pragma unroll
    for (int v = 0; v < 8; ++v) atomicAdd(&s_red[v + hi * 8], p[v]);
  __syncthreads();
  if (tid < 16) s_inv[tid] = 1.0f / fmaxf(sqrtf(s_red[tid]), EPSN);
  __syncthreads();
  float invmn[8];
#pragma unroll
  for (int v = 0; v < 8; ++v) invmn[v] = s_inv[v + hi * 8];
  __syncthreads();
  if (tid < 16) s_red[tid] = 0.0f;
  __syncthreads();

  // ||0.9*mem + 0.1*mean_norm||^2 per class
  float py[8];
#pragma unroll
  for (int v = 0; v < 8; ++v) py[v] = 0.0f;
#pragma unroll
  for (int t = 0; t < 16; ++t)
#pragma unroll
    for (int v = 0; v < 8; ++v) {
      const int gc  = g * 16 + v + hi * 8;
      const int col = wcol + t * 16 + lhalf;
      const float mval = mem[(size_t)gc * D_DIM + col];
      const float y = MOM * mval + (1.0f - MOM) * acc[t][v] * invmn[v];
      py[v] += y * y;
    }
#pragma unroll
  for (int m = 1; m < 16; m <<= 1)
#pragma unroll
    for (int v = 0; v < 8; ++v) py[v] += __shfl_xor(py[v], m, 32);
  if (lhalf == 0)
#pragma unroll
    for (int v = 0; v < 8; ++v) atomicAdd(&s_red[v + hi * 8], py[v]);
  __syncthreads();
  if (tid < 16) s_inv[tid] = 1.0f / fmaxf(sqrtf(s_red[tid]), EPSN);
  __syncthreads();

  // final blended + renormalized write (or passthrough for absent classes)
#pragma unroll
  for (int v = 0; v < 8; ++v) {
    const int   gc    = g * 16 + v + hi * 8;
    const float invbn = s_inv[v + hi * 8];
    const int   pres  = s_pres[v + hi * 8];
#pragma unroll
    for (int t = 0; t < 16; ++t) {
      const int col = wcol + t * 16 + lhalf;
      const float mval = mem[(size_t)gc * D_DIM + col];
      const float y = MOM * mval + (1.0f - MOM) * acc[t][v] * invmn[v];
      outp[(size_t)gc * D_DIM + col] = pres ? (y * invbn) : mval;
    }
  }
}

// ---------------- host-side launch ----------------
extern "C" void kernel_launch(void* const* d_in, const int* in_sizes, int n_in,
                              void* d_out, int out_size, void* d_ws, size_t ws_size,
                              hipStream_t stream)
{
  (void)n_in; (void)out_size; (void)ws_size;
  const float* fv  = (const float*)d_in[0];
  const float* fr  = (const float*)d_in[1];
  const int*   idv = (const int*)d_in[2];
  const int*   idr = (const int*)d_in[3];
  const float* mv  = (const float*)d_in[4];
  const float* mr  = (const float*)d_in[5];
  float* out = (float*)d_out;

  const int N = in_sizes[2];
  const int C = in_sizes[4] / D_DIM;     // 4000

  char* w = (char*)d_ws;
  auto take = [&](size_t bytes) {
    void* p = (void*)w;
    w += (bytes + 255) & ~(size_t)255;
    return p;
  };
  float* invn0 = (float*)take((size_t)N * 4);
  float* invn1 = (float*)take((size_t)N * 4);
  float* wr0   = (float*)take((size_t)N * 4);
  float* wr1   = (float*)take((size_t)N * 4);
  int*   perm0 = (int*)take((size_t)N * 4);
  int*   perm1 = (int*)take((size_t)N * 4);
  int*   cnt0  = (int*)take(CBINS * 4);
  int*   cnt1  = (int*)take(CBINS * 4);
  int*   off0  = (int*)take(CBINS * 4);
  int*   off1  = (int*)take(CBINS * 4);
  int*   cur0  = (int*)take(CBINS * 4);
  int*   cur1  = (int*)take(CBINS * 4);

  k_zero_i32<<<dim3((CBINS + 255) / 256), 256, 0, stream>>>(cnt0, CBINS);
  k_zero_i32<<<dim3((CBINS + 255) / 256), 256, 0, stream>>>(cnt1, CBINS);

  k_rownorm<<<dim3((N + 7) / 8, 2), 256, 0, stream>>>(
      fv, fr, idv, idr, invn0, invn1, cnt0, cnt1, N);

  k_scan<<<dim3(1, 2), 256, 0, stream>>>(cnt0, cnt1, off0, off1, cur0, cur1);

  k_scatter<<<dim3((N + 255) / 256, 2), 256, 0, stream>>>(
      idv, idr, invn0, invn1, cnt0, cnt1, cur0, cur1, perm0, perm1, wr0, wr1, N);

  const int groups = C / 16;             // 250
  k_segmean_wmma<<<groups, 256, 0, stream>>>(
      fv, idv, perm0, wr0, off0, cnt0, mv, out);
  k_segmean_wmma<<<groups, 256, 0, stream>>>(
      fr, idr, perm1, wr1, off1, cnt1, mr, out + (size_t)C * D_DIM);
}